// DeformableConv2D_41609643163909
// MI455X (gfx1250) — compile-verified
//
#include <hip/hip_runtime.h>

typedef __attribute__((ext_vector_type(2))) float v2f;
typedef __attribute__((ext_vector_type(8))) float v8f;

#define Hh 128
#define Ww 128
#define Cc 64
#define Ff 128

// Async copy of 16 bytes: global (per-lane addr) -> LDS (per-lane addr).
// Tracked by ASYNCcnt; async loads complete in order.
static __device__ __forceinline__ void async_b128(unsigned lds_off, const void* gaddr) {
    asm volatile("global_load_async_to_lds_b128 %0, %1, off"
                 :: "v"(lds_off), "v"(gaddr)
                 : "memory");
}
static __device__ __forceinline__ void wait_async9() {   // allow newest 9 in flight
    asm volatile("s_wait_asynccnt 0x9" ::: "memory");
}
static __device__ __forceinline__ void wait_async0() {   // drain
    asm volatile("s_wait_asynccnt 0x0" ::: "memory");
}

// ---------------------------------------------------------------------------
// Kernel A: offset conv (only 2 of 18 output channels are ever used) + tanh +
// truncated/clipped index computation. Writes lin = iy*W + ix per pixel.
// Also writes a 64-float zero page used by kernel B's async gather padding.
// ---------------------------------------------------------------------------
__global__ __launch_bounds__(256)
void deform_offset_idx(const float* __restrict__ x,
                       const float* __restrict__ ow,   // [3,3,64,18]
                       const float* __restrict__ ob,   // [18]
                       int* __restrict__ lin,
                       float* __restrict__ zeropage) {
    if (blockIdx.x == 0 && threadIdx.x < 64) zeropage[threadIdx.x] = 0.0f;

    __shared__ float sw[1152]; // [tap(9)][c(64)][ch(2)]
    for (int i = threadIdx.x; i < 1152; i += 256) {
        int tc = i >> 1;   // tap*64 + c
        int ch = i & 1;
        sw[i] = ow[tc * 18 + ch];
    }
    __syncthreads();

    int id = blockIdx.x * 256 + threadIdx.x;   // pixel id in [0, B*H*W)
    int b   = id >> 14;
    int rem = id & 16383;
    int h   = rem >> 7;
    int w   = rem & 127;

    float s0 = ob[0];
    float s1 = ob[1];
    #pragma unroll
    for (int t = 0; t < 9; ++t) {
        int ky = t / 3, kx = t % 3;
        int y  = h + ky - 1;
        int xp = w + kx - 1;
        if ((unsigned)y < (unsigned)Hh && (unsigned)xp < (unsigned)Ww) {
            const float4* px =
                (const float4*)(x + ((size_t)((b << 14) + (y << 7) + xp) << 6));
            const float* wt = &sw[t * 128];
            #pragma unroll
            for (int c4 = 0; c4 < 16; ++c4) {
                float4 v = px[c4];
                int cb = c4 * 8;
                s0 = fmaf(v.x, wt[cb + 0], s0); s1 = fmaf(v.x, wt[cb + 1], s1);
                s0 = fmaf(v.y, wt[cb + 2], s0); s1 = fmaf(v.y, wt[cb + 3], s1);
                s0 = fmaf(v.z, wt[cb + 4], s0); s1 = fmaf(v.z, wt[cb + 5], s1);
                s0 = fmaf(v.w, wt[cb + 6], s0); s1 = fmaf(v.w, wt[cb + 7], s1);
            }
        }
    }
    float dy = tanhf(s0);
    float dx = tanhf(s1);
    // truncation toward zero (matches astype(int32)), then clip
    int iy = (int)((float)h + dy); iy = min(max(iy, 0), Hh - 1);
    int ix = (int)((float)w + dx); ix = min(max(ix, 0), Ww - 1);
    lin[id] = iy * Ww + ix;
}

// ---------------------------------------------------------------------------
// Kernel B: implicit-GEMM deformable conv using V_WMMA_F32_16X16X4_F32.
// Double-buffered: GLOBAL_LOAD_ASYNC_TO_LDS_B128 stages tap t+1 (weights +
// per-lane gathered deformed rows) while tap t runs on the matrix pipes.
// ---------------------------------------------------------------------------
__global__ __launch_bounds__(256)
void deform_conv_wmma(const float* __restrict__ x,        // [B,H,W,64]
                      const int*   __restrict__ lin,      // [B,H,W]
                      const float* __restrict__ kw,       // [3,3,64,128]
                      const float* __restrict__ kb,       // [128]
                      const float* __restrict__ zeropage, // [64] zeros
                      float* __restrict__ out) {          // [B,H,W,128]
    __shared__ float sA[2][64 * 68];   // gathered pixel rows x 64 ch (pad->68)
    __shared__ float sB[2][64 * 128];  // 64 input ch x 128 out ch

    const int bid  = blockIdx.x;
    const int w0   = (bid & 1) << 6;       // 0 or 64
    const int h    = (bid >> 1) & 127;
    const int b    = bid >> 8;
    const int tid  = threadIdx.x;
    const int lane = tid & 31;
    const int wave = tid >> 5;
    const int fbase = wave << 4;           // this wave's 16 output channels
    const int i16   = lane & 15;
    const int khalf = lane >> 4;

    const size_t xbatch = (size_t)b << 20;  // b * H*W*C

    // Per-thread staging geometry (fixed across taps)
    const int r   = tid >> 2;               // A pixel row 0..63
    const int sub = tid & 3;                // quarter (16 floats) of the row

    // Stage tap tt into buffer s: 8 async B128 for weights + 1 for the gather.
    auto stage = [&](int tt, int s) {
        const float* bsrc = kw + tt * (Cc * Ff);
        #pragma unroll
        for (int i = 0; i < 8; ++i) {
            int idx = tid + i * 256;      // float4 index, 2048 total
            int c   = idx >> 5;           // 32 float4 per 128-f row
            int f4  = idx & 31;
            unsigned ldsb = (unsigned)(uintptr_t)(void*)&sB[s][c * 128 + f4 * 4];
            async_b128(ldsb, bsrc + idx * 4);
        }
        int y  = h + (tt / 3) - 1;
        int xp = w0 + r + (tt % 3) - 1;
        const float* asrc;
        if ((unsigned)y < (unsigned)Hh && (unsigned)xp < (unsigned)Ww) {
            int l = lin[(b << 14) + (y << 7) + xp];
            asrc = x + xbatch + ((size_t)l << 6) + sub * 16;
        } else {
            asrc = zeropage + sub * 16;    // SAME padding -> zeros
        }
        async_b128((unsigned)(uintptr_t)(void*)&sA[s][r * 68 + sub * 16], asrc);
    };

    v8f acc[4];
    #pragma unroll
    for (int m = 0; m < 4; ++m) acc[m] = (v8f){0.f,0.f,0.f,0.f,0.f,0.f,0.f,0.f};

    stage(0, 0);                      // prologue: fill buffer 0

    for (int t = 0; t < 9; ++t) {
        const int s = t & 1;
        __syncthreads();              // all waves done reading buffer s^1
        if (t < 8) {
            stage(t + 1, s ^ 1);      // DMA next tap while we compute this one
            wait_async9();            // oldest 9 (= stage t) have landed
        } else {
            wait_async0();
        }
        __syncthreads();              // everyone's stage-t writes visible

        // ---- compute: 16 k-steps x 4 M-subtiles of f32 WMMA ----
        const float* A = sA[s];
        const float* B = sB[s];
        #pragma unroll
        for (int kk = 0; kk < 64; kk += 4) {
            const int krow = kk + 2 * khalf;
            // B fragment (4x16): lane holds (k=krow,krow+1) at column fbase+i16
            v2f bf;
            bf.x = B[krow * 128 + fbase + i16];
            bf.y = B[(krow + 1) * 128 + fbase + i16];
            #pragma unroll
            for (int m = 0; m < 4; ++m) {
                // A fragment (16x4): lane holds pixel m*16+i16, k=krow..krow+1
                v2f af = *(const v2f*)(&A[(m * 16 + i16) * 68 + krow]);
                acc[m] = __builtin_amdgcn_wmma_f32_16x16x4_f32(
                    false, af, false, bf, (short)0, acc[m], false, false);
            }
        }
    }

    // ---- epilogue: bias + store (C layout: vgpr rr -> M = rr + 8*khalf) ----
    const float kbv = kb[fbase + i16];
    const int pix_base = (b << 14) + (h << 7) + w0;
    #pragma unroll
    for (int m = 0; m < 4; ++m) {
        #pragma unroll
        for (int rr = 0; rr < 8; ++rr) {
            int p = m * 16 + rr + 8 * khalf;
            out[((size_t)(pix_base + p) << 7) + fbase + i16] = acc[m][rr] + kbv;
        }
    }
}

// ---------------------------------------------------------------------------
extern "C" void kernel_launch(void* const* d_in, const int* in_sizes, int n_in,
                              void* d_out, int out_size, void* d_ws, size_t ws_size,
                              hipStream_t stream) {
    const float* x   = (const float*)d_in[0];   // [32,128,128,64]
    const float* owt = (const float*)d_in[1];   // [3,3,64,18]
    const float* obi = (const float*)d_in[2];   // [18]
    const float* kwt = (const float*)d_in[3];   // [3,3,64,128]
    const float* kbi = (const float*)d_in[4];   // [128]
    float* out = (float*)d_out;                 // [32,128,128,128]

    const int npix = 32 * Hh * Ww;              // 524288
    int*   lin      = (int*)d_ws;               // [B*H*W] = 2 MB
    float* zeropage = (float*)((char*)d_ws + (size_t)npix * sizeof(int)); // 256 B

    deform_offset_idx<<<npix / 256, 256, 0, stream>>>(x, owt, obi, lin, zeropage);

    // 8192 blocks: 2 per row (64 px) * 128 rows * 32 batch
    deform_conv_wmma<<<npix / 64, 256, 0, stream>>>(x, lin, kwt, kbi, zeropage, out);
}